// TrajectoryScore_62680752718254
// MI455X (gfx1250) — compile-verified
//
#include <hip/hip_runtime.h>
#include <hip/hip_bf16.h>
#include <math.h>

// ---------------------------------------------------------------------------
// TrajectoryScore for MI455X (gfx1250, wave32)
//
// Roofline: 192 MiB of z streamed once -> ~8.6 us floor at 23.3 TB/s.
// Kernel 1: bandwidth-bound streaming reduction (float4/global_load_b128,
//           v_sqrt_f32 + v_exp_f32, wave32 shfl reduction, deterministic).
// Kernel 2: 64x64 partial matrix row-summed with V_WMMA_F32_16X16X4_F32
//           (A = partials stripe, B = ones) + closed-form moment math.
// ---------------------------------------------------------------------------

#define BATCH            64
#define D1D2             (1024 * 256)            // triples per batch
#define FLOATS_PER_BATCH (D1D2 * 3)              // 786432 floats
#define BLOCKS_PER_BATCH 64
#define TRIPLES_PER_BLK  (D1D2 / BLOCKS_PER_BATCH)   // 4096
#define THREADS          256
#define GROUPS_PER_THR   (TRIPLES_PER_BLK / 4 / THREADS) // 4 (4 triples/group)

typedef float v2f __attribute__((ext_vector_type(2)));
typedef float v8f __attribute__((ext_vector_type(8)));

// ---------------------------------------------------------------------------
// Kernel 1: per-(batch, block) partial sum of exp(-0.5*A*||z||)
// Each thread streams 4 groups of 4 triples = 48 floats via 12 b128 loads.
// ---------------------------------------------------------------------------
__global__ __launch_bounds__(THREADS) void score_partial_kernel(
    const float* __restrict__ z, const float* __restrict__ R,
    float* __restrict__ partial) {
  const int b   = blockIdx.y;
  const int blk = blockIdx.x;
  const int tid = threadIdx.x;

  const float Rb = R[b];
  const float A  = 1.0f / (Rb * Rb);
  // exp(x) = exp2(x*log2e): fold -0.5*A*log2(e) into one coefficient
  const float c2 = -0.5f * A * 1.44269504088896340736f;

  const float4* __restrict__ p = (const float4*)(
      z + (size_t)b * FLOATS_PER_BATCH + (size_t)blk * (TRIPLES_PER_BLK * 3));

  float s = 0.0f;
#pragma unroll
  for (int it = 0; it < GROUPS_PER_THR; ++it) {
    const int g = it * THREADS + tid;  // group of 4 triples = 3 float4s
    float4 q0 = p[g * 3 + 0];
    float4 q1 = p[g * 3 + 1];
    float4 q2 = p[g * 3 + 2];
    float n0 = sqrtf(q0.x * q0.x + q0.y * q0.y + q0.z * q0.z);
    float n1 = sqrtf(q0.w * q0.w + q1.x * q1.x + q1.y * q1.y);
    float n2 = sqrtf(q1.z * q1.z + q1.w * q1.w + q2.x * q2.x);
    float n3 = sqrtf(q2.y * q2.y + q2.z * q2.z + q2.w * q2.w);
    s += __builtin_amdgcn_exp2f(c2 * n0);
    s += __builtin_amdgcn_exp2f(c2 * n1);
    s += __builtin_amdgcn_exp2f(c2 * n2);
    s += __builtin_amdgcn_exp2f(c2 * n3);
  }

  // wave32 butterfly reduction
#pragma unroll
  for (int off = 16; off > 0; off >>= 1) s += __shfl_xor(s, off, 32);

  __shared__ float wsum[THREADS / 32];
  if ((tid & 31) == 0) wsum[tid >> 5] = s;
  __syncthreads();
  if (tid == 0) {
    float t = 0.0f;
#pragma unroll
    for (int w = 0; w < THREADS / 32; ++w) t += wsum[w];  // fixed order
    partial[b * BLOCKS_PER_BATCH + blk] = t;
  }
}

// ---------------------------------------------------------------------------
// i0e(x) for large x (A in [100, 2500] here): asymptotic expansion,
// relative error < 1e-9 for x >= 100.
// ---------------------------------------------------------------------------
__device__ __forceinline__ double i0e_large(double x) {
  const double rx = 1.0 / x;
  const double p =
      1.0 + rx * (0.125 + rx * (0.0703125 + rx * 0.0732421875));
  return p / sqrt(6.283185307179586 * x);
}

// ---------------------------------------------------------------------------
// Kernel 2: row-sum the 64x64 partial matrix with WMMA (P * ones), then
// compute mu / sigma2 / objective. 4 waves, wave w reduces rows 16w..16w+15.
// A-matrix (16x4 f32, 2 VGPRs/lane): lanes 0-15 hold K=0,1; lanes 16-31 K=2,3.
// B = ones(4x16) -> D[m][n] = rowsum(m) in every column (deterministic order).
// ---------------------------------------------------------------------------
__global__ __launch_bounds__(128) void reduce_finalize_kernel(
    const float* __restrict__ P,    // [BATCH][BLOCKS_PER_BATCH]
    const float* __restrict__ R, const int* __restrict__ numObsPtr,
    float* __restrict__ out) {
  __shared__ float rawLds[BATCH];

  const int tid  = threadIdx.x;   // 0..127
  const int wave = tid >> 5;      // 0..3
  const int lane = tid & 31;
  const int base = wave * 16;                 // first batch row of this wave
  const int row  = base + (lane & 15);        // M index
  const int colH = (lane >> 4) << 1;          // 0 for lanes<16, 2 otherwise

  v8f c = {0.f, 0.f, 0.f, 0.f, 0.f, 0.f, 0.f, 0.f};
  v2f bones;
  bones.x = 1.0f;
  bones.y = 1.0f;

#pragma unroll
  for (int k = 0; k < 16; ++k) {              // 16 K-chunks of 4 columns
    const int col = 4 * k + colH;
    v2f a;
    a.x = P[row * BLOCKS_PER_BATCH + col];
    a.y = P[row * BLOCKS_PER_BATCH + col + 1];
    // D = A(16x4) * ones(4x16) + C  => every column of D holds row sums
    c = __builtin_amdgcn_wmma_f32_16x16x4_f32(
        /*neg_a=*/false, a, /*neg_b=*/false, bones,
        /*c_mod=*/(short)0, c, /*reuse_a=*/false, /*reuse_b=*/false);
  }

  // D layout: VGPR r, lanes 0-15 -> M=r ; lanes 16-31 -> M=r+8
  if (lane == 0) {
#pragma unroll
    for (int r = 0; r < 8; ++r) rawLds[base + r] = c[r];
  }
  if (lane == 16) {
#pragma unroll
    for (int r = 0; r < 8; ++r) rawLds[base + 8 + r] = c[r];
  }
  __syncthreads();

  if (tid < BATCH) {
    const double Rb = (double)R[tid];
    const double A  = 1.0 / (Rb * Rb);                 // A in [100, 2500]
    const double e2 = exp(-2.0 * A);
    const double e4 = exp(-4.0 * A);
    const double mean3 = (1.0 - e2) / (2.0 * A);
    const double ms2_3 = (1.0 - e4) / (4.0 * A);
    const double var3  = ms2_3 - mean3 * mean3;
    const double m2d   = i0e_large(A);
    const double v2d   = i0e_large(2.0 * A) - m2d * m2d;
    const double n     = (double)(*numObsPtr);
    const double mu     = n * (0.95 * mean3 + 0.05 * m2d);
    const double sigma2 = n * (0.95 * var3 + 0.05 * v2d);
    const float raw = rawLds[tid];
    const double objective = (double)raw - 1.0 * mu - 0.0 * sigma2;

    out[tid]              = raw;              // raw_score
    out[BATCH + tid]      = (float)mu;        // mu
    out[2 * BATCH + tid]  = (float)sigma2;    // sigma2
    out[3 * BATCH + tid]  = (float)objective; // objective
  }
}

// ---------------------------------------------------------------------------
extern "C" void kernel_launch(void* const* d_in, const int* in_sizes, int n_in,
                              void* d_out, int out_size, void* d_ws,
                              size_t ws_size, hipStream_t stream) {
  const float* z      = (const float*)d_in[0];  // [64,1024,256,3] f32
  const float* R      = (const float*)d_in[1];  // [64] f32
  const int* numObs   = (const int*)d_in[2];    // scalar int
  float* out          = (float*)d_out;          // 256 f32
  float* partial      = (float*)d_ws;           // [64][64] f32 scratch

  dim3 grid(BLOCKS_PER_BATCH, BATCH);
  score_partial_kernel<<<grid, THREADS, 0, stream>>>(z, R, partial);
  reduce_finalize_kernel<<<1, 128, 0, stream>>>(partial, R, numObs, out);
}